// LinearAttentionVarlenModule_84155589198489
// MI455X (gfx1250) — compile-verified
//
#include <hip/hip_runtime.h>

// ---------------------------------------------------------------------------
// Chunked linear attention (varlen) for gfx1250, fp32 WMMA (16x16x4) with
// TDM (tensor_load_to_lds) double-buffered chunk prefetch.
//   per chunk c:  S = tril(Qc Kc^T)  (inclusive diagonal)
//                 O = Qc @ M + S @ Vc
//                 M += Kc^T @ Vc
// One workgroup per packed sequence; 512 threads = 16 wave32 waves.
// ---------------------------------------------------------------------------

typedef __attribute__((ext_vector_type(2))) float        v2f;
typedef __attribute__((ext_vector_type(8))) float        v8f;
typedef __attribute__((ext_vector_type(4))) unsigned int v4u;
typedef __attribute__((ext_vector_type(8))) int          v8i;
typedef __attribute__((ext_vector_type(4))) int          v4i;

#define D        128
#define CHUNK    64
#define NTHREADS 512

// padded row strides (floats) — TDM inserts the +4 pad in hardware
#define QS (D + 4)        // Q/K/V chunk row stride (132)
#define MS (D + 4)        // M row stride
#define SS (CHUNK + 4)    // S row stride (68)

// LDS layout (float offsets)
#define OFF_M   0
#define OFF_S   (OFF_M + D * MS)                 // 16896
#define OFF_BUF (OFF_S + CHUNK * SS)             // 21248
#define BUFSZ   (CHUNK * QS)                     // 8448 floats per Q/K/V tile
#define LDS_FLOATS (OFF_BUF + 6 * BUFSZ)         // 71936 floats = ~281 KB

__device__ __forceinline__ v8f wmma4(v2f a, v2f b, v8f c) {
  return __builtin_amdgcn_wmma_f32_16x16x4_f32(false, a, false, b,
                                               (short)0, c, false, false);
}

// A operand: A[m][k], contiguous k pair -> 64-bit LDS load
__device__ __forceinline__ v2f ldA(const float* p, int stride, int m0, int k0, int lane) {
  const int m = m0 + (lane & 15);
  const int k = k0 + ((lane >> 4) << 1);   // vgpr0 -> k, vgpr1 -> k+1
  return *(const v2f*)(p + m * stride + k);
}
// A operand from transposed source: A[m][k] = src[k][m]
__device__ __forceinline__ v2f ldAT(const float* p, int stride, int m0, int k0, int lane) {
  const int m = m0 + (lane & 15);
  const int k = k0 + ((lane >> 4) << 1);
  v2f r; r.x = p[k * stride + m]; r.y = p[(k + 1) * stride + m]; return r;
}
// B operand: B[k][n], source stored [k][n]
__device__ __forceinline__ v2f ldB(const float* p, int stride, int k0, int n0, int lane) {
  const int n = n0 + (lane & 15);
  const int k = k0 + ((lane >> 4) << 1);
  v2f r; r.x = p[k * stride + n]; r.y = p[(k + 1) * stride + n]; return r;
}
// B operand from transposed source: B[k][n] = src[n][k] (contiguous pair)
__device__ __forceinline__ v2f ldBT(const float* p, int stride, int k0, int n0, int lane) {
  const int n = n0 + (lane & 15);
  const int k = k0 + ((lane >> 4) << 1);
  return *(const v2f*)(p + n * stride + k);
}

// ---------------------------------------------------------------------------
// TDM: async DMA one [rows_avail x 128] f32 tensor tile (tile = CHUNK x 128)
// from global to LDS, padding each 128-DWORD row with 4 DWORDs (stride 132),
// zero-filling rows beyond rows_avail. Tracked by TENSORcnt.
// ---------------------------------------------------------------------------
__device__ __forceinline__ void tdm_load_2d(const float* gptr, const float* ldsptr,
                                            int rows_avail) {
  const unsigned long long ga = (unsigned long long)(uintptr_t)gptr;
  const unsigned lds_byte = (unsigned)(uintptr_t)ldsptr;  // low 32 bits = LDS offset

  v4u g0;
  g0.x = 1u;                                          // count = 1 valid descriptor
  g0.y = lds_byte;                                    // lds_addr [63:32]
  g0.z = (unsigned)(ga & 0xFFFFFFFFu);                // global_addr [95:64]
  g0.w = (unsigned)((ga >> 32) & 0x01FFFFFFu)         // global_addr [120:96]
       | (2u << 30);                                  // type = 2 (image)

  // group1: data_size=4B(2), pad_enable, pad_interval=128 DW(6), pad_amount=4 DW(3)
  const unsigned rows = (unsigned)rows_avail;
  v8i g1;
  g1[0] = (int)((2u << 16) | (1u << 20) | (6u << 22) | (3u << 25));
  g1[1] = (int)(128u << 16);                          // tensor_dim0 = 128
  g1[2] = (int)((rows & 0xFFFFu) << 16);              // tensor_dim1 lo
  g1[3] = (int)((rows >> 16) & 0xFFFFu) | (int)(128u << 16);  // dim1 hi | tile_dim0=128
  g1[4] = (int)CHUNK;                                 // tile_dim1 = 64 (tile_dim2 = 0)
  g1[5] = 128;                                        // tensor_dim0_stride = 128
  g1[6] = 0;                                          // stride hi | tensor_dim1_stride lo
  g1[7] = 0;
  const v4i z4 = {0, 0, 0, 0};
  const v8i z8 = {0, 0, 0, 0, 0, 0, 0, 0};
  // clang-23 / therock-10.0 form: (g0, g1, g2, g3, extra, cpol)
  __builtin_amdgcn_tensor_load_to_lds(g0, g1, z4, z4, z8, 0);
}

// lower-triangular S tile map: 10 tiles over waves 0..9
__device__ __constant__ const signed char S_MB[10] = {0,1,1,2,2,2,3,3,3,3};
__device__ __constant__ const signed char S_NB[10] = {0,0,1,0,1,2,0,1,2,3};

__global__ void __launch_bounds__(NTHREADS)
linattn_varlen_kernel(const float* __restrict__ q, const float* __restrict__ k,
                      const float* __restrict__ v, const int* __restrict__ s,
                      const float* __restrict__ M0, float* __restrict__ out) {
  extern __shared__ float lds[];
  float* Mlds = lds + OFF_M;
  float* Slds = lds + OFF_S;

  const int tid  = threadIdx.x;
  const int lane = tid & 31;
  const int wave = tid >> 5;
  const int wcol = wave & 7;   // d-column tile 0..7 (16 cols each)
  const int wrow = wave >> 3;  // 0..1
  const int cl   = lane & 15;          // D-layout col within tile
  const int rhi  = (lane >> 4) << 3;   // D-layout +8 rows for lanes 16..31

  const int t0 = s[blockIdx.x];
  const int t1 = s[blockIdx.x + 1];

  // ---- M = M_0 ----
  for (int i = tid; i < D * D; i += NTHREADS)
    Mlds[(i >> 7) * MS + (i & 127)] = M0[i];

  // ---- prefetch chunk 0 into buffer 0 (waves 0..2 own q/k/v DMA) ----
  if (wave < 3) {
    const float* src = (wave == 0) ? q : (wave == 1) ? k : v;
    tdm_load_2d(src + (long)t0 * D, lds + OFF_BUF + wave * BUFSZ, t1 - t0);
  }

  int buf = 0;
  for (int c = t0; c < t1; c += CHUNK, buf ^= 1) {
    if (wave < 3) __builtin_amdgcn_s_wait_tensorcnt(0);  // this chunk's DMA done
    __syncthreads();  // chunk visible; prev iter's compute (M,S,K,V reads) done

    // ---- prefetch next chunk into alternate buffer ----
    if (wave < 3 && c + CHUNK < t1) {
      const float* src = (wave == 0) ? q : (wave == 1) ? k : v;
      tdm_load_2d(src + (long)(c + CHUNK) * D,
                  lds + OFF_BUF + ((buf ^ 1) * 3 + wave) * BUFSZ,
                  t1 - (c + CHUNK));
    }

    const float* Qc = lds + OFF_BUF + (buf * 3 + 0) * BUFSZ;
    const float* Kc = lds + OFF_BUF + (buf * 3 + 1) * BUFSZ;
    const float* Vc = lds + OFF_BUF + (buf * 3 + 2) * BUFSZ;

    // ---- Phase 1: S = tril(Q K^T); 10 lower tiles on waves 0..9 ----
    if (wave < 10) {
      const int mb = S_MB[wave], nb = S_NB[wave];
      v8f acc = {};
#pragma unroll
      for (int kb = 0; kb < D; kb += 8) {
        v2f a0 = ldA (Qc, QS, mb * 16, kb,     lane);
        v2f b0 = ldBT(Kc, QS, kb,     nb * 16, lane);
        v2f a1 = ldA (Qc, QS, mb * 16, kb + 4, lane);
        v2f b1 = ldBT(Kc, QS, kb + 4, nb * 16, lane);
        acc = wmma4(a0, b0, acc);
        acc = wmma4(a1, b1, acc);
      }
#pragma unroll
      for (int g = 0; g < 8; ++g) {
        const int rm = mb * 16 + g + rhi;   // query token
        const int cn = nb * 16 + cl;        // key token
        Slds[rm * SS + cn] = (cn <= rm) ? acc[g] : 0.f;
      }
    }
    __syncthreads();

    // ---- Phase 2: O = Q@M + S@V ; wave owns tiles (wrow, wcol),(wrow+2, wcol) ----
#pragma unroll
    for (int ti = 0; ti < 2; ++ti) {
      const int mb = wrow + 2 * ti;
      v8f acc = {};
#pragma unroll
      for (int kb = 0; kb < D; kb += 8) {           // inter-chunk: Q @ M
        v2f a0 = ldA(Qc,   QS, mb * 16, kb,     lane);
        v2f b0 = ldB(Mlds, MS, kb,     wcol * 16, lane);
        v2f a1 = ldA(Qc,   QS, mb * 16, kb + 4, lane);
        v2f b1 = ldB(Mlds, MS, kb + 4, wcol * 16, lane);
        acc = wmma4(a0, b0, acc);
        acc = wmma4(a1, b1, acc);
      }
      for (int nb = 0; nb <= mb; ++nb) {            // intra-chunk: S @ V
#pragma unroll
        for (int j = 0; j < 16; j += 8) {
          v2f a0 = ldA(Slds, SS, mb * 16, nb * 16 + j,     lane);
          v2f b0 = ldB(Vc,   QS, nb * 16 + j,     wcol * 16, lane);
          v2f a1 = ldA(Slds, SS, mb * 16, nb * 16 + j + 4, lane);
          v2f b1 = ldB(Vc,   QS, nb * 16 + j + 4, wcol * 16, lane);
          acc = wmma4(a0, b0, acc);
          acc = wmma4(a1, b1, acc);
        }
      }
#pragma unroll
      for (int g = 0; g < 8; ++g) {
        const int tok = c + mb * 16 + g + rhi;
        if (tok < t1) out[(long)tok * D + wcol * 16 + cl] = acc[g];
      }
    }
    __syncthreads();  // all reads of M done before updating it

    // ---- Phase 3: M += K^T @ V ; wave owns kb in [4*wrow,4*wrow+4), nb=wcol ----
#pragma unroll
    for (int kb4 = 0; kb4 < 4; ++kb4) {
      const int kb = wrow * 4 + kb4;
      v8f acc;
#pragma unroll
      for (int g = 0; g < 8; ++g)
        acc[g] = Mlds[(kb * 16 + g + rhi) * MS + wcol * 16 + cl];
#pragma unroll
      for (int j = 0; j < CHUNK; j += 8) {
        v2f a0 = ldAT(Kc, QS, kb * 16, j,     lane);  // A = K^T (dim x tokens)
        v2f b0 = ldB (Vc, QS, j,       wcol * 16, lane);
        v2f a1 = ldAT(Kc, QS, kb * 16, j + 4, lane);
        v2f b1 = ldB (Vc, QS, j + 4,   wcol * 16, lane);
        acc = wmma4(a0, b0, acc);
        acc = wmma4(a1, b1, acc);
      }
#pragma unroll
      for (int g = 0; g < 8; ++g)
        Mlds[(kb * 16 + g + rhi) * MS + wcol * 16 + cl] = acc[g];
    }
    // no trailing barrier: loop-top barrier orders M writes vs next chunk's reads
  }
}

extern "C" void kernel_launch(void* const* d_in, const int* in_sizes, int n_in,
                              void* d_out, int out_size, void* d_ws, size_t ws_size,
                              hipStream_t stream) {
  const float* q  = (const float*)d_in[0];
  const float* k  = (const float*)d_in[1];
  const float* v  = (const float*)d_in[2];
  const int*   s  = (const int*)d_in[3];
  const float* M0 = (const float*)d_in[4];
  float* out = (float*)d_out;

  const int nseq = in_sizes[3] - 1;  // cu_seqlens has P+1 entries
  const size_t shbytes = (size_t)LDS_FLOATS * sizeof(float);  // ~281 KB < 320 KB/WGP

  linattn_varlen_kernel<<<nseq, NTHREADS, shbytes, stream>>>(q, k, v, s, M0, out);
}